// MambaBlock_47949014893008
// MI455X (gfx1250) — compile-verified
//
#include <hip/hip_runtime.h>
#include <math.h>

#define B_SZ    4
#define L_SZ    2048
#define DM      1024     // D_MODEL
#define DS      16       // D_STATE
#define DC      4        // D_CONV
#define DI      2048     // D_INNER
#define XP_N    33       // 2*D_STATE + 1

typedef __attribute__((ext_vector_type(2))) float v2f;
typedef __attribute__((ext_vector_type(8))) float v8f;

// -----------------------------------------------------------------------------
// C[M,N] = A[M,K] @ W[N,K]^T   (all row-major fp32) via V_WMMA_F32_16X16X4_F32.
// Requires M%256==0, N%128==0, K%8==0, K>=16.
// 256 threads = 8 waves (4x2); each wave computes a 64x64 tile (4x4 fragments).
// Per k-step: 8 b64 fragment loads feed 16 WMMAs; next step's fragments are
// double-buffered in registers so WMMAs overlap the loads.
// A/B fragment layout (32-bit, 16x4): lane L, elem j <-> row (L&15), k 2*(L>>4)+j
// C/D fragment layout: VGPR e, lane L <-> m = e + 8*(L>>4), n = (L&15)
// -----------------------------------------------------------------------------
__global__ __launch_bounds__(256) void gemm_xwt_wmma(
    const float* __restrict__ A, const float* __restrict__ W,
    float* __restrict__ C, int M, int N, int K)
{
  const int lane = threadIdx.x & 31;
  const int wid  = threadIdx.x >> 5;          // 0..7
  const int wm   = wid >> 1;                  // 0..3
  const int wn   = wid & 1;                   // 0..1
  const int row0 = blockIdx.y * 256 + wm * 64;
  const int col0 = blockIdx.x * 128 + wn * 64;
  const int fr   = lane & 15;                 // fragment row (M for A, N for W)
  const int koff = (lane >> 4) << 1;          // 0 or 2

  // 32-bit element offsets (matrices < 2^32 bytes) -> saddr-form global loads.
  unsigned aoff[4], woff[4];
#pragma unroll
  for (int i = 0; i < 4; ++i) aoff[i] = (unsigned)(row0 + i * 16 + fr) * (unsigned)K + koff;
#pragma unroll
  for (int j = 0; j < 4; ++j) woff[j] = (unsigned)(col0 + j * 16 + fr) * (unsigned)K + koff;

  v8f acc[4][4] = {};
  v2f a0[4], b0[4], a1[4], b1[4];

  auto load_frags = [&](v2f (&fa)[4], v2f (&fb)[4], int kk) {
#pragma unroll
    for (int i = 0; i < 4; ++i) fa[i] = *(const v2f*)(A + aoff[i] + kk);
#pragma unroll
    for (int j = 0; j < 4; ++j) fb[j] = *(const v2f*)(W + woff[j] + kk);
  };
  auto do_wmma = [&](v2f (&fa)[4], v2f (&fb)[4]) {
#pragma unroll
    for (int i = 0; i < 4; ++i)
#pragma unroll
      for (int j = 0; j < 4; ++j)
        acc[i][j] = __builtin_amdgcn_wmma_f32_16x16x4_f32(
            false, fa[i], false, fb[j], (short)0, acc[i][j], false, false);
  };

  load_frags(a0, b0, 0);
  int kk = 0;
  for (; kk < K - 8; kk += 8) {
    load_frags(a1, b1, kk + 4);
    do_wmma(a0, b0);
    load_frags(a0, b0, kk + 8);
    do_wmma(a1, b1);
  }
  // tail: kk == K - 8 (no prefetch past end of A/W)
  load_frags(a1, b1, kk + 4);
  do_wmma(a0, b0);
  do_wmma(a1, b1);

  const int cm = (lane >> 4) * 8;
  const int cn = lane & 15;
#pragma unroll
  for (int i = 0; i < 4; ++i)
#pragma unroll
    for (int j = 0; j < 4; ++j) {
      float* Cp = C + (size_t)(row0 + i * 16 + cm) * N + (col0 + j * 16 + cn);
#pragma unroll
      for (int e = 0; e < 8; ++e) Cp[(size_t)e * N] = acc[i][j][e];
    }
}

// -----------------------------------------------------------------------------
// Depthwise causal conv (k=4, left pad 3) + bias + SiLU.
// xz is (B*L, 2*DI) rows; x_h = columns [0,DI). Output xc is (B*L, DI).
// -----------------------------------------------------------------------------
__global__ __launch_bounds__(256) void conv_silu(
    const float* __restrict__ xz, const float* __restrict__ cw,
    const float* __restrict__ cb, float* __restrict__ xc)
{
  const size_t idx = (size_t)blockIdx.x * 256 + threadIdx.x;
  const size_t total = (size_t)B_SZ * L_SZ * DI;
  if (idx >= total) return;
  const int d = (int)(idx % DI);
  const int l = (int)((idx / DI) % L_SZ);
  const int b = (int)(idx / ((size_t)DI * L_SZ));
  const size_t rowb = (size_t)b * L_SZ * (2 * DI);

  float s = cb[d];
#pragma unroll
  for (int t = 0; t < DC; ++t) {
    const int ll = l + t - (DC - 1);
    if (ll >= 0) s += cw[d * DC + t] * xz[rowb + (size_t)ll * (2 * DI) + d];
  }
  const float sv = s / (1.f + expf(-s));   // SiLU
  xc[((size_t)b * L_SZ + l) * DI + d] = sv;
}

// -----------------------------------------------------------------------------
// xp[row, j] = sum_d xc[row, d] * W_x[j, d],  j = 0..32.  One block per row.
// -----------------------------------------------------------------------------
__global__ __launch_bounds__(256) void xproj(
    const float* __restrict__ xc, const float* __restrict__ Wx,
    float* __restrict__ xp)
{
  __shared__ float sx[DI];
  __shared__ float red[256];
  const int tid = threadIdx.x;
  const size_t row = blockIdx.x;
  const float* xr = xc + row * DI;
#pragma unroll
  for (int i = 0; i < DI / 256; ++i) sx[tid + 256 * i] = xr[tid + 256 * i];
  __syncthreads();

  for (int j = 0; j < XP_N; ++j) {
    const float* w = Wx + (size_t)j * DI;
    float p = 0.f;
#pragma unroll
    for (int i = 0; i < DI / 256; ++i) p += sx[tid + 256 * i] * w[tid + 256 * i];
    red[tid] = p;
    __syncthreads();
    for (int s = 128; s > 0; s >>= 1) {
      if (tid < s) red[tid] += red[tid + s];
      __syncthreads();
    }
    if (tid == 0) xp[row * XP_N + j] = red[0];
    __syncthreads();
  }
}

// -----------------------------------------------------------------------------
// Selective scan: one thread per (b, d) channel, 16 states in registers,
// sequential over L. xp row (dt_raw | B[16] | C[16]) staged in LDS per step.
// Emits y * silu(z) directly.
// -----------------------------------------------------------------------------
__global__ __launch_bounds__(256) void scan_kernel(
    const float* __restrict__ xc, const float* __restrict__ xp,
    const float* __restrict__ xz, const float* __restrict__ W_dt,
    const float* __restrict__ b_dt, const float* __restrict__ A_log,
    const float* __restrict__ D_par, float* __restrict__ y)
{
  const int tid = threadIdx.x;
  const int b = blockIdx.x >> 3;                       // 4 batches
  const int d = ((blockIdx.x & 7) << 8) + tid;         // 8 blocks x 256 = 2048

  float Arow[DS];
#pragma unroll
  for (int n = 0; n < DS; ++n) Arow[n] = -expf(A_log[d * DS + n]);
  const float wdt = W_dt[d];
  const float bdt = b_dt[d];
  const float Dp  = D_par[d];

  float h[DS] = {};
  __shared__ float sxp[XP_N];
  const size_t rowbase = (size_t)b * L_SZ;

  for (int l = 0; l < L_SZ; ++l) {
    const size_t row = rowbase + l;
    if (tid < XP_N) sxp[tid] = xp[row * XP_N + tid];
    __syncthreads();

    const float dtr = sxp[0];
    const float sp  = (dtr > 20.f) ? dtr : log1pf(expf(dtr));   // softplus
    const float dt  = sp * wdt + bdt;
    const float x_t = xc[row * DI + d];
    const float z_t = xz[row * (2 * DI) + DI + d];
    const float dtx = dt * x_t;

    float yt = 0.f;
#pragma unroll
    for (int n = 0; n < DS; ++n) {
      const float dA = expf(dt * Arow[n]);
      h[n] = dA * h[n] + dtx * sxp[1 + n];
      yt += h[n] * sxp[1 + DS + n];
    }
    yt += Dp * x_t;
    const float sz = z_t / (1.f + expf(-z_t));
    y[row * DI + d] = yt * sz;
    __syncthreads();
  }
}

// -----------------------------------------------------------------------------
extern "C" void kernel_launch(void* const* d_in, const int* in_sizes, int n_in,
                              void* d_out, int out_size, void* d_ws, size_t ws_size,
                              hipStream_t stream)
{
  const float* x      = (const float*)d_in[0];
  const float* W_in   = (const float*)d_in[1];
  const float* conv_w = (const float*)d_in[2];
  const float* conv_b = (const float*)d_in[3];
  const float* W_x    = (const float*)d_in[4];
  const float* W_dt   = (const float*)d_in[5];
  const float* b_dt   = (const float*)d_in[6];
  const float* A_log  = (const float*)d_in[7];
  const float* D_par  = (const float*)d_in[8];
  const float* W_out  = (const float*)d_in[9];
  float* out = (float*)d_out;

  const size_t M = (size_t)B_SZ * L_SZ;   // 8192 rows

  float* ws = (float*)d_ws;
  float* xz = ws;                         // M * 4096
  float* xc = xz + M * (2 * DI);          // M * 2048
  float* xp = xc + M * DI;                // M * 33
  float* ym = xp + M * XP_N;              // M * 2048

  dim3 blk(256);

  // 1) xz = x @ W_in^T   (8192 x 4096, K = 1024)
  gemm_xwt_wmma<<<dim3((2 * DI) / 128, M / 256), blk, 0, stream>>>(
      x, W_in, xz, (int)M, 2 * DI, DM);

  // 2) depthwise causal conv + SiLU -> xc
  const size_t totalc = M * DI;
  conv_silu<<<dim3((unsigned)((totalc + 255) / 256)), blk, 0, stream>>>(
      xz, conv_w, conv_b, xc);

  // 3) xp = xc @ W_x^T   (N = 33)
  xproj<<<dim3((unsigned)M), blk, 0, stream>>>(xc, W_x, xp);

  // 4) selective scan + gating -> ym
  scan_kernel<<<dim3(B_SZ * (DI / 256)), blk, 0, stream>>>(
      xc, xp, xz, W_dt, b_dt, A_log, D_par, ym);

  // 5) out = ym @ W_out^T   (8192 x 1024, K = 2048)
  gemm_xwt_wmma<<<dim3(DM / 128, M / 256), blk, 0, stream>>>(
      ym, W_out, out, (int)M, DM, DI);
}